// RefFrameCalc_30202210026025
// MI455X (gfx1250) — compile-verified
//
#include <hip/hip_runtime.h>

// RefFrameCalc: per-edge 3-vector frame construction.
// Memory-bound streaming kernel: 132 B/edge * 4M edges = 528 MB once-through
// => ~23 us floor at 23.3 TB/s. Strategy: 4 edges per thread so all global
// traffic is 16B-aligned b128, non-temporal (don't churn the 192MB L2).
// VALU trimmed with v_rsq_f32 (exact clamp semantics preserved) so compute
// stays comfortably under the memory roof.

typedef float v4f __attribute__((ext_vector_type(4)));

struct F3 { float x, y, z; };

__device__ __forceinline__ F3 f3add(F3 a, F3 b) { return {a.x + b.x, a.y + b.y, a.z + b.z}; }
__device__ __forceinline__ F3 f3sub(F3 a, F3 b) { return {a.x - b.x, a.y - b.y, a.z - b.z}; }
__device__ __forceinline__ F3 f3scale(F3 a, float s) { return {a.x * s, a.y * s, a.z * s}; }
__device__ __forceinline__ float f3dot(F3 a, F3 b) { return fmaf(a.x, b.x, fmaf(a.y, b.y, a.z * b.z)); }
__device__ __forceinline__ F3 f3cross(F3 a, F3 b) {
    return {fmaf(a.y, b.z, -a.z * b.y),
            fmaf(a.z, b.x, -a.x * b.z),
            fmaf(a.x, b.y, -a.y * b.x)};
}

// 1 / max(sqrt(d2), 1e-6)  ==  rsq(d2) if d2 > 1e-12 else 1e6
// (single v_rsq_f32 + v_cndmask instead of sqrt-fixup + rcp-Newton chains)
__device__ __forceinline__ float inv_norm_clamped(float d2) {
    float r = __builtin_amdgcn_rsqf(d2);
    return (d2 > 1e-12f) ? r : 1e6f;
}

// normalize with clamped norm (matches reference _normalize, EPS=1e-6)
__device__ __forceinline__ F3 f3normeps(F3 t) {
    return f3scale(t, inv_norm_clamped(f3dot(t, t)));
}

__device__ __forceinline__ void compute_edge(
    F3 SP, F3 RP, F3 SV, F3 RV, F3 SPV, F3 RPV, F3 SO, F3 RO,
    F3& A, F3& B, F3& C)
{
    F3 rel = f3sub(RP, SP);
    // vector_a = rel / sqrt(dot)  (no clamp, matches reference)
    F3 a = f3scale(rel, __builtin_amdgcn_rsqf(f3dot(rel, rel)));

    F3 diff_vel  = f3sub(RV, SV);
    F3 sum_vel   = f3add(SV, RV);
    F3 diff_pvel = f3sub(RPV, SPV);
    F3 sum_pvel  = f3add(SPV, RPV);
    F3 diff_om   = f3sub(RO, SO);
    F3 sum_om    = f3add(SO, RO);

    F3 b_i   = f3normeps(f3cross(diff_vel, a));
    F3 b_ii  = f3normeps(sum_vel);
    F3 b_iii = f3normeps(f3cross(diff_om, a));
    F3 b_iv  = f3normeps(sum_om);
    F3 b_v   = f3normeps(f3cross(diff_pvel, a));
    F3 b_vi  = f3normeps(sum_pvel);

    F3 b = f3add(f3add(f3add(b_i, b_ii), f3add(b_iii, b_iv)), f3add(b_v, b_vi));
    b = {b.x + 1.0f, b.y + 1.0f, b.z + 1.0f};

    F3 b_prl = f3scale(a, f3dot(b, a));
    F3 b_prp = f3sub(b, b_prl);
    F3 vb = f3normeps(f3cross(b_prp, a));
    F3 vc = f3normeps(f3cross(b_prl, vb));

    A = a; B = vb; C = vc;
}

__device__ __forceinline__ void store12_nt(float* __restrict__ dst, const float* src) {
    v4f w0 = {src[0], src[1], src[2],  src[3]};
    v4f w1 = {src[4], src[5], src[6],  src[7]};
    v4f w2 = {src[8], src[9], src[10], src[11]};
    v4f* p = reinterpret_cast<v4f*>(dst);
    __builtin_nontemporal_store(w0, p + 0);
    __builtin_nontemporal_store(w1, p + 1);
    __builtin_nontemporal_store(w2, p + 2);
}

__global__ __launch_bounds__(256) void ref_frame_kernel(
    const float* __restrict__ sp,  const float* __restrict__ rp,
    const float* __restrict__ sv,  const float* __restrict__ rv,
    const float* __restrict__ spv, const float* __restrict__ rpv,
    const float* __restrict__ so,  const float* __restrict__ ro,
    float* __restrict__ out, long long E)
{
    const long long gid  = (long long)blockIdx.x * blockDim.x + threadIdx.x;
    const long long base = gid * 4;                 // 4 edges per thread
    if (base >= E) return;

    const long long secB = 3LL * E;                 // start of vector_b section
    const long long secC = 6LL * E;                 // start of vector_c section
    const float* __restrict__ ptrs[8] = {sp, rp, sv, rv, spv, rpv, so, ro};

    const bool vec_ok = (base + 4 <= E) && ((E & 3LL) == 0LL); // 16B-aligned sections

    if (vec_ok) {
        // ---- fast path: 24 NT b128 loads, 9 NT b128 stores, all 16B aligned ----
        float in[8][12];
        const long long base3 = base * 3;           // byte offset 48*gid -> 16B aligned
#pragma unroll
        for (int k = 0; k < 8; ++k) {
            const v4f* p = reinterpret_cast<const v4f*>(ptrs[k] + base3);
            v4f w0 = __builtin_nontemporal_load(p + 0);
            v4f w1 = __builtin_nontemporal_load(p + 1);
            v4f w2 = __builtin_nontemporal_load(p + 2);
            in[k][0] = w0[0]; in[k][1] = w0[1]; in[k][2]  = w0[2]; in[k][3]  = w0[3];
            in[k][4] = w1[0]; in[k][5] = w1[1]; in[k][6]  = w1[2]; in[k][7]  = w1[3];
            in[k][8] = w2[0]; in[k][9] = w2[1]; in[k][10] = w2[2]; in[k][11] = w2[3];
        }

        float oa[12], ob[12], oc[12];
#pragma unroll
        for (int j = 0; j < 4; ++j) {
            const int o = 3 * j;
            F3 A, B, C;
            compute_edge({in[0][o], in[0][o+1], in[0][o+2]},
                         {in[1][o], in[1][o+1], in[1][o+2]},
                         {in[2][o], in[2][o+1], in[2][o+2]},
                         {in[3][o], in[3][o+1], in[3][o+2]},
                         {in[4][o], in[4][o+1], in[4][o+2]},
                         {in[5][o], in[5][o+1], in[5][o+2]},
                         {in[6][o], in[6][o+1], in[6][o+2]},
                         {in[7][o], in[7][o+1], in[7][o+2]},
                         A, B, C);
            oa[o] = A.x; oa[o+1] = A.y; oa[o+2] = A.z;
            ob[o] = B.x; ob[o+1] = B.y; ob[o+2] = B.z;
            oc[o] = C.x; oc[o+1] = C.y; oc[o+2] = C.z;
        }

        store12_nt(out +        base3, oa);
        store12_nt(out + secB + base3, ob);
        store12_nt(out + secC + base3, oc);
    } else {
        // ---- tail path: scalar per edge ----
        const long long end = (base + 4 < E) ? (base + 4) : E;
        for (long long e = base; e < end; ++e) {
            F3 v[8];
#pragma unroll
            for (int k = 0; k < 8; ++k) {
                const float* p = ptrs[k] + e * 3;
                v[k] = {p[0], p[1], p[2]};
            }
            F3 A, B, C;
            compute_edge(v[0], v[1], v[2], v[3], v[4], v[5], v[6], v[7], A, B, C);
            float* pa = out +        e * 3;
            float* pb = out + secB + e * 3;
            float* pc = out + secC + e * 3;
            pa[0] = A.x; pa[1] = A.y; pa[2] = A.z;
            pb[0] = B.x; pb[1] = B.y; pb[2] = B.z;
            pc[0] = C.x; pc[1] = C.y; pc[2] = C.z;
        }
    }
}

extern "C" void kernel_launch(void* const* d_in, const int* in_sizes, int n_in,
                              void* d_out, int out_size, void* d_ws, size_t ws_size,
                              hipStream_t stream) {
    // setup_inputs order:
    // 0: edge_index (int64 [2,E]) -- UNUSED by the reference computation
    // 1: senders_pos  2: receivers_pos  3: senders_vel  4: receivers_vel
    // 5: senders_prev_vel  6: receivers_prev_vel  7: senders_omega  8: receivers_omega
    const float* sp  = (const float*)d_in[1];
    const float* rp  = (const float*)d_in[2];
    const float* sv  = (const float*)d_in[3];
    const float* rv  = (const float*)d_in[4];
    const float* spv = (const float*)d_in[5];
    const float* rpv = (const float*)d_in[6];
    const float* so  = (const float*)d_in[7];
    const float* ro  = (const float*)d_in[8];
    float* out = (float*)d_out;

    const long long E = (long long)in_sizes[1] / 3;   // [E,3] float32
    const long long threads = (E + 3) / 4;            // 4 edges per thread
    const int block = 256;                            // 8 wave32 waves per block
    const int grid = (int)((threads + block - 1) / block);

    ref_frame_kernel<<<grid, block, 0, stream>>>(sp, rp, sv, rv, spv, rpv, so, ro, out, E);
}